// AffinityMatrixConstructLayer_43490838839881
// MI455X (gfx1250) — compile-verified
//
#include <hip/hip_runtime.h>
#include <math.h>

// ---------------------------------------------------------------------------
// AffinityMatrixConstructLayer for MI455X (gfx1250, wave32, WMMA)
//
// Sizes: n1=n2=48, e1=e2=192, gd=dn=de=1024. Output M is 2304x2304 f32.
// Bandwidth-bound (~40MB traffic); GEMMs done with V_WMMA_F32_16X16X4_F32
// to keep exact f32 numerics while using the matrix pipe.
// ---------------------------------------------------------------------------

typedef __attribute__((ext_vector_type(2))) float v2f;
typedef __attribute__((ext_vector_type(8))) float v8f;

#define GD 1024
#define N1 48
#define E1 192
#define MDIM (N1 * N1)   // 2304

// ---- coeff = tanh(W @ gw + b), one wave32 per output row --------------------
__global__ void coeff_tanh_matvec(const float* __restrict__ W,
                                  const float* __restrict__ b,
                                  const float* __restrict__ gw,
                                  float* __restrict__ out, int rows) {
  int wavesPerBlock = blockDim.x >> 5;
  int wid  = blockIdx.x * wavesPerBlock + (threadIdx.x >> 5);
  int lane = threadIdx.x & 31;
  if (wid >= rows) return;
  const float* row = W + (size_t)wid * GD;
  float s = 0.f;
#pragma unroll 4
  for (int k = lane; k < GD; k += 32) s += row[k] * gw[k];
#pragma unroll
  for (int off = 16; off > 0; off >>= 1) s += __shfl_down(s, off, 32);
  if (lane == 0) out[wid] = tanhf(s + b[wid]);
}

// ---- relu(softplus(x) - 0.5), numerically stable softplus -------------------
__device__ __forceinline__ float softplus_relu_half(float x) {
  float sp = fmaxf(x, 0.f) + log1pf(expf(-fabsf(x)));
  return fmaxf(sp - 0.5f, 0.f);
}

// ---- Out[i,j] = f( sum_k A[i,k]*coeff[k]*B[j,k] ), one wave per 16x16 tile --
// Uses V_WMMA_F32_16X16X4_F32: A is 16x4 (M in lanes, K split lo/hi half-wave),
// B is 4x16 (N in lanes, same K split) -> symmetric load pattern.
__global__ void affinity_wmma(const float* __restrict__ A,
                              const float* __restrict__ B,
                              const float* __restrict__ coeff,
                              float* __restrict__ Out, int ldo) {
  const int lane  = threadIdx.x;        // 0..31, full wave active (EXEC all 1s)
  const int idx16 = lane & 15;
  const int koff  = (lane >> 4) << 1;   // 0 for lanes 0-15, 2 for lanes 16-31

  const float* arow = A + (size_t)(blockIdx.x * 16 + idx16) * GD;
  const float* brow = B + (size_t)(blockIdx.y * 16 + idx16) * GD;

  v8f acc = {};
  for (int k0 = 0; k0 < GD; k0 += 4) {
    const int k = k0 + koff;
    const float2 cv = *(const float2*)(coeff + k);
    const float2 av = *(const float2*)(arow + k);
    const float2 bv = *(const float2*)(brow + k);
    v2f a, b;
    a.x = av.x * cv.x;  a.y = av.y * cv.y;   // fold coeff into A (applied once)
    b.x = bv.x;         b.y = bv.y;
    acc = __builtin_amdgcn_wmma_f32_16x16x4_f32(
        /*neg_a=*/false, a, /*neg_b=*/false, b,
        /*c_mod=*/(short)0, acc, /*reuse_a=*/false, /*reuse_b=*/false);
  }

  // C/D layout: VGPR v -> row (v + 8*(lane>=16)), col = lane%16
  const int n     = idx16;
  const int mBase = (lane >> 4) << 3;   // 0 or 8
#pragma unroll
  for (int v = 0; v < 8; ++v) {
    const int m = blockIdx.x * 16 + mBase + v;
    Out[(size_t)m * ldo + blockIdx.y * 16 + n] = softplus_relu_half(acc[v]);
  }
}

// ---- M = diag(vec_Mp) (zero elsewhere); vec_Mp[r] = Mp row-major flat[r] ----
__global__ void assemble_diag(const float* __restrict__ Mp,
                              float* __restrict__ M) {
  const int total = MDIM * MDIM;
  int idx = blockIdx.x * blockDim.x + threadIdx.x;
  if (idx >= total) return;
  int r = idx / MDIM;
  int c = idx - r * MDIM;
  M[idx] = (r == c) ? Mp[r] : 0.f;
}

// ---- Kg @ diag(vec_Me) @ Kh^T collapses to scatter-add over edge pairs -----
// kron col = b2*e1 + b1  ->  vec_Me = Me_flat[b2*192 + b1]
// row r = src2*48 + src1, col c = dst2*48 + dst1 (one 1 per incidence column)
__global__ void scatter_me(const float* __restrict__ Me,
                           const int* __restrict__ ei1,
                           const int* __restrict__ ei2,
                           float* __restrict__ M) {
  int idx = blockIdx.x * blockDim.x + threadIdx.x;
  if (idx >= E1 * E1) return;
  int b2 = idx / E1;
  int b1 = idx - b2 * E1;
  int src2 = ei2[b2], dst2 = ei2[E1 + b2];   // edge_index2[0,b2], [1,b2]
  int src1 = ei1[b1], dst1 = ei1[E1 + b1];   // edge_index1[0,b1], [1,b1]
  int r = src2 * N1 + src1;
  int c = dst2 * N1 + dst1;
  atomicAdd(M + (size_t)r * MDIM + c, Me[idx]);  // duplicate edges possible
}

extern "C" void kernel_launch(void* const* d_in, const int* in_sizes, int n_in,
                              void* d_out, int out_size, void* d_ws, size_t ws_size,
                              hipStream_t stream) {
  (void)in_sizes; (void)n_in; (void)out_size; (void)ws_size;

  const float* x1  = (const float*)d_in[0];
  const float* x2  = (const float*)d_in[1];
  const float* ef1 = (const float*)d_in[2];
  const float* ef2 = (const float*)d_in[3];
  const float* gw  = (const float*)d_in[4];
  const float* Wn  = (const float*)d_in[5];
  const float* bn  = (const float*)d_in[6];
  const float* We  = (const float*)d_in[7];
  const float* be  = (const float*)d_in[8];
  const int*   ei1 = (const int*)d_in[9];
  const int*   ei2 = (const int*)d_in[10];
  float* M = (float*)d_out;

  float* ws      = (float*)d_ws;
  float* coeff_n = ws;            // 1024 floats
  float* coeff_e = ws + 1024;     // 1024 floats
  float* Mp      = ws + 2048;     // 48*48   = 2304 floats
  float* Me      = ws + 8192;     // 192*192 = 36864 floats

  // 1) coeff vectors: 8 waves per block, one row per wave
  coeff_tanh_matvec<<<128, 256, 0, stream>>>(Wn, bn, gw, coeff_n, GD);
  coeff_tanh_matvec<<<128, 256, 0, stream>>>(We, be, gw, coeff_e, GD);

  // 2) WMMA affinity GEMMs: one wave32 per 16x16 output tile
  dim3 gMp(N1 / 16, N1 / 16);    // 3x3 tiles
  dim3 gMe(E1 / 16, E1 / 16);    // 12x12 tiles
  affinity_wmma<<<gMp, 32, 0, stream>>>(x1,  x2,  coeff_n, Mp, N1);
  affinity_wmma<<<gMe, 32, 0, stream>>>(ef1, ef2, coeff_e, Me, E1);

  // 3) stream-write M = diag(vec_Mp)
  assemble_diag<<<(MDIM * MDIM + 255) / 256, 256, 0, stream>>>(Mp, M);

  // 4) sparse scatter of Me through the (implicit) Kronecker incidence
  scatter_me<<<(E1 * E1 + 255) / 256, 256, 0, stream>>>(Me, ei1, ei2, M);
}